// ParabolicIntegrate_16037407883385
// MI455X (gfx1250) — compile-verified
//
#include <hip/hip_runtime.h>
#include <hip/hip_bf16.h>

// ---------------------------------------------------------------------------
// ParabolicIntegrate on MI455X (gfx1250, wave32, WMMA).
//
// Sizes: B=128, NT=200, NX=256, 12 graph features. Recurrences
//   y_t = (y_{t-1} + dt*tmp_t) @ M,  M = inv(I - c A).T  (256x256 f32)
// are computed with V_WMMA_F32_16X16X4_F32, keeping the 256KB B-matrix
// resident in LDS for the whole 199-step scan (CDNA5: 320KB LDS per WGP).
//
// d_ws requirement: 2 * 256*256 * 4 bytes = 512 KiB (two in-place inverses).
// ---------------------------------------------------------------------------

#define B_    128
#define NT_   200
#define NX_   256
#define NG_   12
#define DT_F  (1.0f / 199.0f)
// c = eps*dt/dx^2 with dx = 1/255
#define CC_F  ((1.0f / 199.0f) * 255.0f * 255.0f)

typedef __attribute__((ext_vector_type(2))) float v2f;
typedef __attribute__((ext_vector_type(8))) float v8f;

// ---------------------------------------------------------------------------
// A-matrix entry of the discrete Laplacian with BC fixups (mirrors reference).
// bc == 0 : periodic ('P'), bc == 1 : Dirichlet ('D')
// ---------------------------------------------------------------------------
__device__ __forceinline__ float Aval(int i, int j, int bc) {
    const int N = NX_ - 1;
    float v = 0.0f;
    if (i == j) v = -2.0f;
    else if (j == i + 1 || j == i - 1) v = 1.0f;
    if (bc) {  // Dirichlet: zero the corner couplings
        if ((i == 0 && (j == 0 || j == 1)) || (i == 1 && j == 0)) v = 0.0f;
        if ((i == N && (j == N || j == N - 1)) || (i == N - 1 && j == N)) v = 0.0f;
    } else {   // Periodic: A[-1,1]=1, A[0,-2]=1
        if (i == N && j == 1)     v = 1.0f;
        if (i == 0 && j == N - 1) v = 1.0f;
    }
    return v;
}

// ---------------------------------------------------------------------------
// In-place Gauss-Jordan inverse of (I - c*A), no pivoting (strictly diagonally
// dominant: |1+2c| > 2|c|). One block per matrix: blockIdx 0 -> periodic (M),
// blockIdx 1 -> Dirichlet (M_c). Result: ws[bc][n][k] = P[n][k] = inv(...)[n][k].
// ---------------------------------------------------------------------------
__global__ void __launch_bounds__(256) gj_invert_kernel(float* __restrict__ ws) {
    const int bc  = blockIdx.x;
    float* a      = ws + (size_t)bc * NX_ * NX_;
    const int tid = threadIdx.x;
    __shared__ float prow[NX_];

    // init: a = I - c*A   (thread = row)
    for (int j = 0; j < NX_; ++j)
        a[tid * NX_ + j] = ((tid == j) ? 1.0f : 0.0f) - CC_F * Aval(tid, j, bc);
    __syncthreads();

    for (int p = 0; p < NX_; ++p) {
        const float pv = a[p * NX_ + p];
        __syncthreads();
        // scale pivot row; pivot column entry is replaced by 1 before scaling
        {
            const float ipv = 1.0f / pv;
            const float val = ((tid == p) ? 1.0f : a[p * NX_ + tid]) * ipv;
            prow[tid]        = val;
            a[p * NX_ + tid] = val;
        }
        __syncthreads();
        if (tid != p) {
            const float dum = a[tid * NX_ + p];
            for (int j = 0; j < NX_; ++j) {
                const float cur = (j == p) ? 0.0f : a[tid * NX_ + j];
                a[tid * NX_ + j] = cur - dum * prow[j];
            }
        }
        __syncthreads();
    }
}

// ---------------------------------------------------------------------------
// Feature 0: dW[:,0]=0 ; dW[:,t]=(W[:,t]-W[:,t-1])/dt
// ---------------------------------------------------------------------------
__global__ void __launch_bounds__(256) dw_kernel(const float* __restrict__ W,
                                                 float* __restrict__ out) {
    const int idx = blockIdx.x * 256 + threadIdx.x;   // over B*NT*NX
    const int t   = (idx >> 8) % NT_;
    float v = 0.0f;
    if (t > 0) v = (W[idx] - W[idx - NX_]) * (1.0f / DT_F);
    out[(size_t)idx * NG_ + 0] = v;
}

// ---------------------------------------------------------------------------
// Time-scan kernel. One block = 16 batch rows x full NX, 8 wave32s.
// stage 0: ic  (y_0 = U0, y_t = y_{t-1} @ M_c), node 1
// stage 1: nodes {2,3,4,5,10}  (parents among feats 0,1)
// stage 2: nodes {6,7,8,9,11}  (parents among feats 0,1,2,4)
//   y_0 = 0, y_t = (y_{t-1} + dt * prod feats[p]^q) @ M
// B-matrix held in LDS as interleaved K-pairs: Mb[kk*256+n] = {B[2kk][n],B[2kk+1][n]}
// with B[k][n] = P[n][k].
// ---------------------------------------------------------------------------
__global__ void __launch_bounds__(256) scan_kernel(const float* __restrict__ Pinv,
                                                   const float* __restrict__ U0,
                                                   float* __restrict__ out,
                                                   int stage) {
    __shared__ v2f   Mb[128 * NX_];   // 256 KB: B operand, K-pair interleaved
    __shared__ float ab[16 * NX_];    //  16 KB: A operand / running y

    static const int NODE[3][5] = {{1, 0, 0, 0, 0}, {2, 3, 4, 5, 10}, {6, 7, 8, 9, 11}};
    static const int P0T[3][5]  = {{0, 0, 0, 0, 0}, {0, 1, 0, 0, 1},  {2, 2, 2, 0, 4}};
    static const int Q0T[3][5]  = {{1, 1, 1, 1, 1}, {1, 1, 2, 1, 2},  {2, 1, 3, 1, 1}};
    static const int P1T[3][5]  = {{-1,-1,-1,-1,-1},{-1,-1,-1, 1,-1}, {-1, 1,-1, 2, 2}};
    static const int Q1T[3][5]  = {{1, 1, 1, 1, 1}, {1, 1, 1, 1, 1},  {1, 1, 1, 1, 1}};

    const int tid  = threadIdx.x;
    const int wave = tid >> 5;
    const int lane = tid & 31;
    const int half = lane >> 4;   // K sub-tile / C row-half select
    const int l15  = lane & 15;
    const int b0   = blockIdx.x * 16;
    const int sel  = blockIdx.y;
    const int node = NODE[stage][sel];
    const int p0   = P0T[stage][sel], q0 = Q0T[stage][sel];
    const int p1   = P1T[stage][sel], q1 = Q1T[stage][sel];
    const bool ic  = (stage == 0);

    // --- stage B-matrix into LDS (coalesced per-row float2 copies) ---------
    for (int j = 0; j < 32; ++j) {
        const int n = wave + 8 * j;
        const v2f* src = (const v2f*)(Pinv + (size_t)n * NX_);  // row n of P
#pragma unroll
        for (int q = 0; q < 4; ++q) {
            const int kk = lane + 32 * q;
            Mb[kk * NX_ + n] = src[kk];
        }
    }

    // --- init y_0 and emit t=0 output --------------------------------------
    for (int i = 0; i < 16; ++i) {
        const int e = tid + 256 * i;
        const int row = e >> 8, col = e & 255;
        const float v = ic ? U0[(b0 + row) * NX_ + col] : 0.0f;
        ab[e] = v;
        out[((size_t)((b0 + row) * NT_ + 0) * NX_ + col) * NG_ + node] = v;
    }
    __syncthreads();

    const int n0 = wave * 32;   // each wave owns columns [n0, n0+32)

    for (int t = 1; t < NT_; ++t) {
        // phase 1: A = y_{t-1} + dt * tmp_t (graph product read from out)
        if (!ic) {
#pragma unroll 4
            for (int i = 0; i < 16; ++i) {
                const int e = tid + 256 * i;
                const int row = e >> 8, col = e & 255;
                const float* fb =
                    out + ((size_t)((b0 + row) * NT_ + t) * NX_ + col) * NG_;
                const float f0 = fb[p0];
                float v = f0;
                if (q0 >= 2) v *= f0;
                if (q0 == 3) v *= f0;
                if (p1 >= 0) {
                    const float f1 = fb[p1];
                    float w = f1;
                    if (q1 >= 2) w *= f1;
                    v *= w;
                }
                ab[e] += DT_F * v;
            }
        }
        __syncthreads();

        // phase 2: y_t = A @ B via f32 WMMA, K = 256 in 64 steps of 4
        v8f c0 = {0.f, 0.f, 0.f, 0.f, 0.f, 0.f, 0.f, 0.f};
        v8f c1 = c0;
        const float* arow = ab + l15 * NX_;   // A row = lane&15
#pragma unroll 8
        for (int kk = 0; kk < 128; kk += 2) {
            const int ks = kk + half;         // K pair index for this lane-half
            v2f a;                             // A[m][k0+2h], A[m][k0+2h+1]
            a.x = arow[2 * ks];
            a.y = arow[2 * ks + 1];
            const v2f bb0 = Mb[ks * NX_ + n0 + l15];        // B k-pair, N-tile 0
            const v2f bb1 = Mb[ks * NX_ + n0 + 16 + l15];   // B k-pair, N-tile 1
            c0 = __builtin_amdgcn_wmma_f32_16x16x4_f32(false, a, false, bb0,
                                                       (short)0, c0, false, false);
            c1 = __builtin_amdgcn_wmma_f32_16x16x4_f32(false, a, false, bb1,
                                                       (short)0, c1, false, false);
        }
        __syncthreads();

        // phase 3: write y_t back to LDS (becomes next A) and to global out
#pragma unroll
        for (int v = 0; v < 8; ++v) {
            const int row = v + 8 * half;     // C layout: VGPR v, lane-half
            const int ca = n0 + l15, cb = n0 + 16 + l15;
            ab[row * NX_ + ca] = c0[v];
            ab[row * NX_ + cb] = c1[v];
            const size_t ob =
                ((size_t)((b0 + row) * NT_ + t) * NX_) * NG_ + node;
            out[ob + (size_t)ca * NG_] = c0[v];
            out[ob + (size_t)cb * NG_] = c1[v];
        }
        __syncthreads();
    }
}

// ---------------------------------------------------------------------------
extern "C" void kernel_launch(void* const* d_in, const int* in_sizes, int n_in,
                              void* d_out, int out_size, void* d_ws, size_t ws_size,
                              hipStream_t stream) {
    (void)in_sizes; (void)n_in; (void)out_size; (void)ws_size;
    const float* W  = (const float*)d_in[0];
    const float* U0 = (const float*)d_in[1];
    float* out = (float*)d_out;
    float* ws  = (float*)d_ws;   // needs >= 512 KiB

    // 1) build P = inv(I - cA) for periodic (M) and Dirichlet (M_c)
    gj_invert_kernel<<<2, 256, 0, stream>>>(ws);
    // 2) feature 0 (dW)
    dw_kernel<<<(B_ * NT_ * NX_) / 256, 256, 0, stream>>>(W, out);
    // 3) feature 1 (ic scan, Dirichlet matrix)
    scan_kernel<<<dim3(8, 1), 256, 0, stream>>>(ws + (size_t)NX_ * NX_, U0, out, 0);
    // 4) stage-1 nodes {2,3,4,5,10} (periodic matrix)
    scan_kernel<<<dim3(8, 5), 256, 0, stream>>>(ws, U0, out, 1);
    // 5) stage-2 nodes {6,7,8,9,11}
    scan_kernel<<<dim3(8, 5), 256, 0, stream>>>(ws, U0, out, 2);
}